// BoxCrossAttention_352187318473
// MI455X (gfx1250) — compile-verified
//
#include <hip/hip_runtime.h>
#include <math.h>

typedef float v2f __attribute__((ext_vector_type(2)));
typedef float v8f __attribute__((ext_vector_type(8)));

__device__ __forceinline__ float mish_f(float v) {
    // mish(v) = v * tanh(softplus(v)); guard exp overflow (softplus(v)≈v for v>20)
    float sp = (v > 20.0f) ? v : log1pf(expf(v));
    return v * tanhf(sp);
}

// Out[0..3, tile*16 + 0..15] = act(A[4,K] @ B[:, bcol0 + tile*16 + j] + bias) * scale
// One wave32 per 16-column tile using V_WMMA_F32_16X16X4_F32 (fp32-native matrix op).
// A rows 4..15 of the 16x4 fragment are duplicates of rows 0..3 (in-bounds loads);
// they only affect D rows 4..15, which are never stored.
__global__ __launch_bounds__(32)
void gemm4_wmma_kernel(const float* __restrict__ A, int K,
                       const float* __restrict__ B, int ldb, int bcol0,
                       const float* __restrict__ bias,
                       float* __restrict__ Out, int ldo,
                       int act, float scale)
{
    const int lane = threadIdx.x & 31;
    const int half = lane >> 4;        // 0 -> K lanes k0+0,k0+1 ; 1 -> k0+2,k0+3
    const int lm   = lane & 15;
    const int arow = lm & 3;           // fold 16 M-rows onto the 4 real batch rows
    const int ncol = bcol0 + blockIdx.x * 16 + lm;
    const int koff = half * 2;

    v8f acc = {};
    #pragma unroll 4
    for (int k0 = 0; k0 < K; k0 += 4) {
        v2f a, b;
        a.x = A[arow * K + (k0 + koff)];
        a.y = A[arow * K + (k0 + koff + 1)];
        b.x = B[(k0 + koff)     * ldb + ncol];
        b.y = B[(k0 + koff + 1) * ldb + ncol];
        acc = __builtin_amdgcn_wmma_f32_16x16x4_f32(false, a, false, b,
                                                    (short)0, acc, false, false);
    }

    // C/D layout: lanes 0..15 hold N=lane, VGPR r = row M=r (r=0..7).
    if (half == 0) {
        const int oc = blockIdx.x * 16 + lm;
        const float bb = bias[oc];
        #pragma unroll
        for (int r = 0; r < 4; ++r) {
            float v = acc[r] + bb;
            if (act) v = mish_f(v);
            Out[r * ldo + oc] = v * scale;
        }
    }
}

// out[plane, :] = ovec[plane], plane = b*256 + c, 4096 floats per plane.
// 16 MB of b128 stores -> pure HBM-store-bandwidth bound (~0.7 us at 23.3 TB/s).
__global__ __launch_bounds__(256)
void broadcast_plane_kernel(const float* __restrict__ ovec, float* __restrict__ out)
{
    const int plane = blockIdx.x;                 // uniform -> scalar load
    const float v = ovec[plane];
    const float4 vv = make_float4(v, v, v, v);
    float4* dst = (float4*)(out + (size_t)plane * 4096);
    const int t = threadIdx.x;
    #pragma unroll
    for (int i = 0; i < 4; ++i)
        dst[i * 256 + t] = vv;                    // global_store_b128
}

extern "C" void kernel_launch(void* const* d_in, const int* in_sizes, int n_in,
                              void* d_out, int out_size, void* d_ws, size_t ws_size,
                              hipStream_t stream)
{
    (void)in_sizes; (void)n_in; (void)out_size; (void)ws_size;

    // setup_inputs() order:
    // 0:x 1:y 2:Wp 3:bp 4:W1 5:b1 6:W2 7:b2 8:Wq 9:bq 10:Wk 11:bk 12:Wv 13:bv 14:Wo 15:bo
    const float* y  = (const float*)d_in[1];
    const float* W1 = (const float*)d_in[4];
    const float* b1 = (const float*)d_in[5];
    const float* W2 = (const float*)d_in[6];
    const float* b2 = (const float*)d_in[7];
    const float* Wv = (const float*)d_in[12];
    const float* bv = (const float*)d_in[13];
    const float* Wo = (const float*)d_in[14];
    const float* bo = (const float*)d_in[15];
    float* out = (float*)d_out;

    float* ws   = (float*)d_ws;
    float* t1   = ws;                 // [4,1024] mish(y@W1+b1)
    float* vraw = t1   + 4 * 1024;    // [4,512]  v half of kv
    float* vp   = vraw + 4 * 512;     // [4,256]
    float* ovec = vp   + 4 * 256;     // [4,256]  9*(vp@Wo+bo)

    // 1) t1 = mish(y @ W1 + b1)                  (64 tiles of N=1024)
    gemm4_wmma_kernel<<<64, 32, 0, stream>>>(y, 256, W1, 1024, 0, b1,
                                             t1, 1024, /*act=*/1, 1.0f);
    // 2) vraw = t1 @ W2[:, 512:1024] + b2[512:]  (32 tiles of N=512)
    gemm4_wmma_kernel<<<32, 32, 0, stream>>>(t1, 1024, W2, 1024, 512, b2 + 512,
                                             vraw, 512, /*act=*/0, 1.0f);
    // 3) vp = vraw @ Wv + bv                     (16 tiles of N=256)
    gemm4_wmma_kernel<<<16, 32, 0, stream>>>(vraw, 512, Wv, 256, 0, bv,
                                             vp, 256, /*act=*/0, 1.0f);
    // 4) ovec = 9 * (vp @ Wo + bo)               (16 tiles of N=256)
    gemm4_wmma_kernel<<<16, 32, 0, stream>>>(vp, 256, Wo, 256, 0, bo,
                                             ovec, 256, /*act=*/0, 9.0f);
    // 5) out[b,c,:,:] = ovec[b,c]  (softmax over length-1 key == 1 -> broadcast)
    broadcast_plane_kernel<<<4 * 256, 256, 0, stream>>>(ovec, out);
}